// OpinionStateCost_32701880991913
// MI455X (gfx1250) — compile-verified
//
#include <hip/hip_runtime.h>
#include <hip/hip_bf16.h>

// Problem sizes (fixed by the reference)
#define BB 32
#define NN 64
#define TT 64
#define DD 256
#define KC 128          // K chunk per staging phase (2 phases cover D=256)
#define RS 132          // LDS row stride in floats: 128 + 4 pad -> 64-bank conflict-free

typedef __attribute__((ext_vector_type(2))) float v2f;
typedef __attribute__((ext_vector_type(4))) float v4f;
typedef __attribute__((ext_vector_type(8))) float v8f;
typedef __attribute__((ext_vector_type(4))) int   i4;

__device__ __forceinline__ float lane_shfl(float v, int src) {
  return __shfl(v, src, 32);
}
__device__ __forceinline__ float lane_shfl_xor(float v, int mask) {
  return __shfl_xor(v, mask, 32);
}

// 16-byte global -> LDS copy. Prefer the CDNA5 async-DMA path (ASYNCcnt).
__device__ __forceinline__ void cp16(const float* g, float* l) {
#if __has_builtin(__builtin_amdgcn_global_load_async_to_lds_b128)
  __builtin_amdgcn_global_load_async_to_lds_b128(
      (__attribute__((address_space(1))) i4*)((i4*)g),
      (__attribute__((address_space(3))) i4*)((i4*)l),
      /*imm offset*/ 0, /*cpol*/ 0);
#else
  *(v4f*)l = *(const v4f*)g;
#endif
}

__device__ __forceinline__ void wait_async() {
#if __has_builtin(__builtin_amdgcn_global_load_async_to_lds_b128)
  asm volatile("s_wait_asynccnt 0x0" ::: "memory");
#endif
}

// One wave computes a 16-row x 32-col tile of xmu for one t via fp32 WMMA.
// Per block: 8 waves share one t-slice of mean_t (staged once into LDS) and
// each stages its own 16xKC xt tile via async global->LDS.
__global__ __launch_bounds__(256) void opinion_state_cost_kernel(
    const float* __restrict__ xt,      // (B,N,T,D)
    const float* __restrict__ mean_t,  // (B,T,D)
    const float* __restrict__ gamma_t, // (B,T,1)
    float* __restrict__ out)           // (B,N,T)
{
  constexpr float LOG_2PI = 1.8378770664093453f;
  constexpr float LOG_B   = 3.4657359027997265f; // log(32)

  extern __shared__ float smem[];          // [32*RS] mean | [8*16*RS] xt tiles
  float* msh = smem;
  float* xsh = smem + 32 * RS;

  const int tid  = threadIdx.x;
  const int lane = tid & 31;
  const int wave = tid >> 5;
  const int t    = blockIdx.x >> 4;                 // 64 time slices
  const int tile = (blockIdx.x & 15) * 8 + wave;    // 128 row tiles of 16
  const int half = lane >> 4;                       // 0: lanes 0-15, 1: lanes 16-31
  const int l15  = lane & 15;
  const int rowBase = tile * 16;

  float* xw = xsh + wave * 16 * RS;                 // this wave's xt tile

  v8f acc0 = {};
  v8f acc1 = {};
  float x2p = 0.f, mu2p0 = 0.f, mu2p1 = 0.f;

  for (int p = 0; p < 2; ++p) {
    const int k0 = p * KC;

    // Stage mean_t[:, t, k0:k0+KC): 32 rows x 32 chunks of 16B, whole block.
    for (int i = tid; i < 32 * (KC / 4); i += 256) {
      const int c  = i >> 5;           // row (mixture component)
      const int d4 = i & 31;           // 16B chunk within row
      cp16(mean_t + ((size_t)c * TT + t) * DD + k0 + d4 * 4,
           msh + c * RS + d4 * 4);
    }
    // Stage this wave's xt tile: 16 rows x 32 chunks of 16B, coalesced.
    for (int i = lane; i < 16 * (KC / 4); i += 32) {
      const int r  = i >> 5;
      const int d4 = i & 31;
      cp16(xt + ((size_t)(rowBase + r) * TT + t) * DD + k0 + d4 * 4,
           xw + r * RS + d4 * 4);
    }
    wait_async();
    __syncthreads();

    // Fragment pointers (16x4 f32 A layout: lane holds row l15, K pair 2*half).
    const float* ap  = xw  + l15 * RS + 2 * half;
    const float* b0p = msh + l15 * RS + 2 * half;          // cols 0-15
    const float* b1p = msh + (l15 + 16) * RS + 2 * half;   // cols 16-31

#pragma unroll 8
    for (int k = 0; k < KC; k += 4) {
      v2f a  = *(const v2f*)(ap  + k);
      v2f b0 = *(const v2f*)(b0p + k);
      v2f b1 = *(const v2f*)(b1p + k);

      // ||x||^2 and ||mu||^2 partials ride along in the VALU
      x2p   += a.x  * a.x  + a.y  * a.y;
      mu2p0 += b0.x * b0.x + b0.y * b0.y;
      mu2p1 += b1.x * b1.x + b1.y * b1.y;

      // D = A x B + C on the fp32 matrix core (exact vs. reference)
      acc0 = __builtin_amdgcn_wmma_f32_16x16x4_f32(
          false, a, false, b0, (short)0, acc0, false, false);
      acc1 = __builtin_amdgcn_wmma_f32_16x16x4_f32(
          false, a, false, b1, (short)0, acc1, false, false);
    }
    __syncthreads();   // protect LDS before next phase restages
  }

  // Each half-wave saw K = {2*half, 2*half+1} mod 4 -> combine halves.
  x2p   += lane_shfl_xor(x2p, 16);
  mu2p0 += lane_shfl_xor(mu2p0, 16);
  mu2p1 += lane_shfl_xor(mu2p1, 16);
  // Now: lane holds x2[rowBase + l15], mu2[l15], mu2[l15+16].

  // Per-column Gaussian constants. gamma_t flat index = c*T + t.
  const int c0 = l15, c1 = l15 + 16;
  const float g0 = gamma_t[c0 * TT + t];
  const float g1 = gamma_t[c1 * TT + t];
  const float inv0 = 1.f / (g0 * g0);
  const float inv1 = 1.f / (g1 * g1);
  const float cst0 = -(float)DD * __logf(g0) - 0.5f * (float)DD * LOG_2PI;
  const float cst1 = -(float)DD * __logf(g1) - 0.5f * (float)DD * LOG_2PI;

  // C/D layout: VGPR v holds row M = v + 8*half for column N = l15 (+16 for acc1).
#pragma unroll
  for (int v = 0; v < 8; ++v) {
    const int rowLocal = v + 8 * half;
    const float x2row = lane_shfl(x2p, rowLocal);  // x2 lives at lane == rowLocal

    const float e0 = -0.5f * (x2row - 2.f * acc0[v] + mu2p0) * inv0 + cst0;
    const float e1 = -0.5f * (x2row - 2.f * acc1[v] + mu2p1) * inv1 + cst1;

    // Pairwise LSE of this lane's two columns
    float m = fmaxf(e0, e1);
    float s = __expf(e0 - m) + __expf(e1 - m);

    // LSE-merge across the 16 lanes of this half (masks stay within the half)
#pragma unroll
    for (int mask = 1; mask <= 8; mask <<= 1) {
      const float om = lane_shfl_xor(m, mask);
      const float os = lane_shfl_xor(s, mask);
      const float nm = fmaxf(m, om);
      s = s * __expf(m - nm) + os * __expf(om - nm);
      m = nm;
    }

    if (l15 == 0) {  // lanes 0 and 16 each own one row
      const int row = rowBase + rowLocal;          // flattened b*N + n
      out[(size_t)row * TT + t] = m + __logf(s) - LOG_B;  // ENT == 1.0
    }
  }
}

extern "C" void kernel_launch(void* const* d_in, const int* in_sizes, int n_in,
                              void* d_out, int out_size, void* d_ws, size_t ws_size,
                              hipStream_t stream) {
  (void)in_sizes; (void)n_in; (void)out_size; (void)d_ws; (void)ws_size;
  const float* xt      = (const float*)d_in[0];  // (B,N,T,D)
  // d_in[1] is `t`, unused by the computation
  const float* mean_t  = (const float*)d_in[2];  // (B,T,D)
  const float* gamma_t = (const float*)d_in[3];  // (B,T,1)
  float* out = (float*)d_out;                    // (B,N,T)

  // 64 t-slices * 16 tile groups; 8 waves (one row-tile each) per block.
  dim3 grid(TT * 16);
  dim3 block(256);
  const size_t shmem = (size_t)(32 * RS + 8 * 16 * RS) * sizeof(float); // 84480 B
  opinion_state_cost_kernel<<<grid, block, shmem, stream>>>(xt, mean_t, gamma_t, out);
}